// ODEfunc_78804059947257
// MI455X (gfx1250) — compile-verified
//
#include <hip/hip_runtime.h>
#include <hip/hip_bf16.h>

// CNF ODE func (dz/dt + exact -tr(J)) fully fused for gfx1250.
//
// tr(J) identity: J = W0z D1 W1z D2 W2z  =>  tr(J) = s1^T (W1z .* (W2z W0z)^T) s2
// so the O(B*32*256*256) jacfwd collapses to ONE extra [B,256]@[256,256] GEMM.
// All GEMMs use native f32 WMMA (v_wmma_f32_16x16x4_f32) => exact f32 math.

typedef __attribute__((ext_vector_type(2))) float v2f;
typedef __attribute__((ext_vector_type(8))) float v8f;

#define HIDDEN 256
#define LDS_LD 260            // padded row stride (dwords) -> conflict-free K reads
#define DIN 32
#define BATCH 16384
#define WAVES 4
#define ROWS_PER_WAVE 16
#define ROWS_PER_BLOCK (WAVES * ROWS_PER_WAVE)   // 64

__device__ __forceinline__ v8f wmma_f32(v2f a, v2f b, v8f c) {
  // D = A(16x4,f32) * B(4x16,f32) + C(16x16,f32)
  return __builtin_amdgcn_wmma_f32_16x16x4_f32(false, a, false, b, (short)0, c,
                                               false, false);
}

__device__ __forceinline__ v8f splat8(float x) {
  v8f v = {x, x, x, x, x, x, x, x};
  return v;
}

__device__ __forceinline__ void lds_fence() {
  // per-wave LDS RAW fence between phases (DS is in-order; this also stops
  // the compiler from reordering across the phase boundary)
  asm volatile("s_wait_dscnt 0" ::: "memory");
}

__device__ __forceinline__ float softplus_f(float x) {
  // logaddexp(x, 0) = max(x,0) + log1p(exp(-|x|))  (matches jax.nn.softplus)
  return fmaxf(x, 0.0f) + log1pf(expf(-fabsf(x)));
}

__device__ __forceinline__ float sigmoid_f(float x) {
  return 1.0f / (1.0f + expf(-x));
}

// ---------------------------------------------------------------------------
// G[j,k] = W1[1+j, k] * M[k, j],  M = W2z @ W0z   (constant 256x256, ~4 MFLOP)
// ---------------------------------------------------------------------------
__global__ void build_g_kernel(const float* __restrict__ W0,
                               const float* __restrict__ W1,
                               const float* __restrict__ W2,
                               float* __restrict__ G) {
  const int j = blockIdx.x;    // 0..255  (s1 index, K-dim of V gemm)
  const int k = threadIdx.x;   // 0..255  (s2 index, N-dim of V gemm)
  float m = 0.0f;
  #pragma unroll
  for (int i = 0; i < DIN; ++i)
    m += W2[(1 + k) * DIN + i] * W0[(1 + i) * HIDDEN + j];
  G[j * HIDDEN + k] = W1[(1 + j) * HIDDEN + k] * m;
}

// ---------------------------------------------------------------------------
// Fused forward + trace kernel. One wave handles 16 batch rows end-to-end.
// ---------------------------------------------------------------------------
__global__ __launch_bounds__(WAVES * 32) void cnf_fused_kernel(
    const float* __restrict__ z, const float* __restrict__ t,
    const float* __restrict__ W0, const float* __restrict__ b0,
    const float* __restrict__ W1, const float* __restrict__ b1,
    const float* __restrict__ W2, const float* __restrict__ b2,
    const float* __restrict__ G, float* __restrict__ dz_out,
    float* __restrict__ dlogp_out) {
  extern __shared__ float smem[];

  const int lane = threadIdx.x & 31;
  const int wv   = threadIdx.x >> 5;
  const int ln   = lane & 15;    // column/row-within-16 index
  const int hs   = lane >> 4;    // lane-half select (K pair / M+8 select)

  float* cur = smem + wv * (ROWS_PER_WAVE * LDS_LD);                      // h
  float* sig = smem + (WAVES + wv) * (ROWS_PER_WAVE * LDS_LD);            // s1 -> V

  const int mb = blockIdx.x * ROWS_PER_BLOCK + wv * ROWS_PER_WAVE;
  const float tv = t[0];

  v8f acc[16];

  // ================= Layer 0:  pre1 = [t|z] @ W0 + b0  (K=32, N=256) ========
  v2f a0[8];
  {
    const float* zr = z + (size_t)(mb + ln) * DIN + 2 * hs;
    #pragma unroll
    for (int kk = 0; kk < 8; ++kk)
      a0[kk] = *(const v2f*)(zr + 4 * kk);
  }
  #pragma unroll
  for (int j = 0; j < 16; ++j) {
    const int n = j * 16 + ln;
    acc[j] = splat8(b0[n] + tv * W0[n]);   // W0 row 0 = time channel
  }
  #pragma unroll
  for (int kk = 0; kk < 8; ++kk) {
    const float* wr = W0 + (size_t)(1 + 4 * kk + 2 * hs) * HIDDEN;
    #pragma unroll
    for (int j = 0; j < 16; ++j) {
      const int n = j * 16 + ln;
      v2f bf = {wr[n], wr[HIDDEN + n]};
      acc[j] = wmma_f32(a0[kk], bf, acc[j]);
    }
  }
  // epilogue: h1 -> cur, s1 -> sig
  #pragma unroll
  for (int j = 0; j < 16; ++j) {
    const int n = j * 16 + ln;
    #pragma unroll
    for (int r = 0; r < 8; ++r) {
      const int row = r + 8 * hs;
      const float x = acc[j][r];
      cur[row * LDS_LD + n] = softplus_f(x);
      sig[row * LDS_LD + n] = sigmoid_f(x);
    }
  }
  lds_fence();

  // ================= V = S1 @ G   (K=256, N=256)  [the trace GEMM] ==========
  #pragma unroll
  for (int j = 0; j < 16; ++j) acc[j] = splat8(0.0f);
  #pragma unroll 1
  for (int kk = 0; kk < 64; ++kk) {
    v2f a = *(const v2f*)(sig + ln * LDS_LD + 4 * kk + 2 * hs);
    const float* gr = G + (size_t)(4 * kk + 2 * hs) * HIDDEN;
    #pragma unroll
    for (int j = 0; j < 16; ++j) {
      const int n = j * 16 + ln;
      v2f bf = {gr[n], gr[HIDDEN + n]};
      acc[j] = wmma_f32(a, bf, acc[j]);
    }
  }
  lds_fence();
  // overwrite sig with V (s1 fully consumed)
  #pragma unroll
  for (int j = 0; j < 16; ++j) {
    const int n = j * 16 + ln;
    #pragma unroll
    for (int r = 0; r < 8; ++r)
      sig[(r + 8 * hs) * LDS_LD + n] = acc[j][r];
  }
  lds_fence();

  // ================= Layer 1:  pre2 = [t|h1] @ W1 + b1  (K=256, N=256) ======
  #pragma unroll
  for (int j = 0; j < 16; ++j) {
    const int n = j * 16 + ln;
    acc[j] = splat8(b1[n] + tv * W1[n]);
  }
  #pragma unroll 1
  for (int kk = 0; kk < 64; ++kk) {
    v2f a = *(const v2f*)(cur + ln * LDS_LD + 4 * kk + 2 * hs);
    const float* wr = W1 + (size_t)(1 + 4 * kk + 2 * hs) * HIDDEN;
    #pragma unroll
    for (int j = 0; j < 16; ++j) {
      const int n = j * 16 + ln;
      v2f bf = {wr[n], wr[HIDDEN + n]};
      acc[j] = wmma_f32(a, bf, acc[j]);
    }
  }
  lds_fence();
  // epilogue: h2 -> cur (over h1), trace partial = sum_k sigmoid(pre2)*V
  float tacc[8];
  #pragma unroll
  for (int r = 0; r < 8; ++r) tacc[r] = 0.0f;
  #pragma unroll
  for (int j = 0; j < 16; ++j) {
    const int n = j * 16 + ln;
    #pragma unroll
    for (int r = 0; r < 8; ++r) {
      const int row = r + 8 * hs;
      const float x = acc[j][r];
      cur[row * LDS_LD + n] = softplus_f(x);
      tacc[r] += sigmoid_f(x) * sig[row * LDS_LD + n];
    }
  }
  // reduce each row's partial across its 16 lanes (xor butterfly stays in half)
  #pragma unroll
  for (int r = 0; r < 8; ++r) {
    tacc[r] += __shfl_xor(tacc[r], 1, 32);
    tacc[r] += __shfl_xor(tacc[r], 2, 32);
    tacc[r] += __shfl_xor(tacc[r], 4, 32);
    tacc[r] += __shfl_xor(tacc[r], 8, 32);
  }
  if (ln == 0) {
    #pragma unroll
    for (int r = 0; r < 8; ++r)
      dlogp_out[mb + r + 8 * hs] = -tacc[r];
  }
  lds_fence();

  // ================= Layer 2:  out = [t|h2] @ W2 + b2  (K=256, N=32) ========
  v8f oacc[2];
  #pragma unroll
  for (int j = 0; j < 2; ++j) {
    const int n = j * 16 + ln;
    oacc[j] = splat8(b2[n] + tv * W2[n]);
  }
  #pragma unroll 1
  for (int kk = 0; kk < 64; ++kk) {
    v2f a = *(const v2f*)(cur + ln * LDS_LD + 4 * kk + 2 * hs);
    const float* wr = W2 + (size_t)(1 + 4 * kk + 2 * hs) * DIN;
    #pragma unroll
    for (int j = 0; j < 2; ++j) {
      const int n = j * 16 + ln;
      v2f bf = {wr[n], wr[DIN + n]};
      oacc[j] = wmma_f32(a, bf, oacc[j]);
    }
  }
  #pragma unroll
  for (int j = 0; j < 2; ++j) {
    const int n = j * 16 + ln;
    #pragma unroll
    for (int r = 0; r < 8; ++r)
      dz_out[(size_t)(mb + r + 8 * hs) * DIN + n] = oacc[j][r];
  }
}

// ---------------------------------------------------------------------------
extern "C" void kernel_launch(void* const* d_in, const int* in_sizes, int n_in,
                              void* d_out, int out_size, void* d_ws,
                              size_t ws_size, hipStream_t stream) {
  const float* z  = (const float*)d_in[0];
  const float* t  = (const float*)d_in[1];
  const float* W0 = (const float*)d_in[2];
  const float* b0 = (const float*)d_in[3];
  const float* W1 = (const float*)d_in[4];
  const float* b1 = (const float*)d_in[5];
  const float* W2 = (const float*)d_in[6];
  const float* b2 = (const float*)d_in[7];

  float* out       = (float*)d_out;          // dz_dt: [B,32]
  float* dlogp     = out + (size_t)BATCH * DIN;  // then dlogp: [B,1]
  float* G         = (float*)d_ws;           // 256*256 f32 = 256 KB scratch

  build_g_kernel<<<HIDDEN, HIDDEN, 0, stream>>>(W0, W1, W2, G);

  const size_t smem_bytes = 2ull * WAVES * ROWS_PER_WAVE * LDS_LD * sizeof(float);
  cnf_fused_kernel<<<BATCH / ROWS_PER_BLOCK, WAVES * 32, smem_bytes, stream>>>(
      z, t, W0, b0, W1, b1, W2, b2, G, out, dlogp);
}